// LIF_18863496364595
// MI455X (gfx1250) — compile-verified
//
#include <hip/hip_runtime.h>
#include <cstdint>

// LIF forward scan, [B=32, T=16, C=128, H=32, W=32] f32.
// Memory-bound (537 MB @ 23.3 TB/s ~= 23 us). Strategy:
//  - 1 thread per (b, chw/4) site, float4 (B128) granularity
//  - 8-deep async global->LDS ring prefetch (ASYNCcnt) over the T loop,
//    NT temporal hint (read-once 256 MiB stream > 192 MB L2)
//  - nontemporal B128 stores for the once-written output stream

typedef __attribute__((ext_vector_type(4))) float v4f;

#define LIF_B      32
#define LIF_T      16
#define LIF_CHW4   32768          // (128*32*32)/4 float4 per (b,t)
#define LIF_TPB    256
#define STAGES     8
#define STAGE_BYTES (LIF_TPB * 16)   // 4 KB per stage -> 32 KB ring

__global__ __launch_bounds__(LIF_TPB) void lif_fwd_kernel(
    const v4f* __restrict__ x, v4f* __restrict__ o)
{
  __shared__ v4f stage[STAGES][LIF_TPB];   // 32 KB ring buffer

  const int    tid   = threadIdx.x;
  const int    b     = blockIdx.x >> 7;    // 128 blocks per batch
  const int    chunk = blockIdx.x & 127;
  const size_t i     = (size_t)chunk * LIF_TPB + (size_t)tid;
  const size_t base  = (size_t)b * ((size_t)LIF_T * LIF_CHW4) + i;
  const v4f*   xp    = x + base;
  v4f*         op    = o + base;

  // Raw LDS byte offset of the ring (flat LDS addr low 32 bits = group offset).
  const unsigned lds_base = (unsigned)(uintptr_t)(&stage[0][0]);
  const unsigned slot     = (unsigned)tid * 16u;

  // Prime the ring: async-DMA timesteps 0..7 into LDS (ASYNCcnt += 8).
  #pragma unroll
  for (int s = 0; s < STAGES; ++s) {
    asm volatile("global_load_async_to_lds_b128 %0, %1, off th:TH_LOAD_NT"
                 :: "v"(lds_base + (unsigned)s * STAGE_BYTES + slot),
                    "v"(xp + (size_t)s * LIF_CHW4)
                 : "memory");
  }

  v4f u = {0.f, 0.f, 0.f, 0.f};   // membrane potential

  // Per step t: allow min(STAGES-1, 15-t) younger async loads outstanding ->
  // load t has completed (per-wave async loads complete in order). Consume
  // from LDS, compute, NT-store spikes, then prefetch t+STAGES into the slot.
  #define LIF_STEP(t)                                                          \
  {                                                                            \
    asm volatile("s_wait_asynccnt %0"                                          \
                 :: "i"(((15 - (t)) < (STAGES - 1)) ? (15 - (t)) : (STAGES - 1)) \
                 : "memory");                                                  \
    v4f xt = stage[(t) & (STAGES - 1)][tid];                                   \
    v4f sp;                                                                    \
    _Pragma("unroll")                                                          \
    for (int k = 0; k < 4; ++k) {                                              \
      u[k]  = __builtin_fmaf(0.5f, u[k], xt[k]);  /* leaky integrate   */      \
      sp[k] = (u[k] >= 1.0f) ? 1.0f : 0.0f;       /* spike (u-1 >= 0)  */      \
      u[k] -= sp[k];                              /* soft reset        */      \
    }                                                                          \
    __builtin_nontemporal_store(sp, op + (size_t)(t) * LIF_CHW4);              \
    if ((t) + STAGES < LIF_T) {                                                \
      asm volatile("global_load_async_to_lds_b128 %0, %1, off th:TH_LOAD_NT"   \
                   :: "v"(lds_base +                                           \
                          (unsigned)(((t) + STAGES) & (STAGES - 1)) * STAGE_BYTES + slot), \
                      "v"(xp + (size_t)((t) + STAGES) * LIF_CHW4)              \
                   : "memory");                                                \
    }                                                                          \
  }

  LIF_STEP(0)  LIF_STEP(1)  LIF_STEP(2)  LIF_STEP(3)
  LIF_STEP(4)  LIF_STEP(5)  LIF_STEP(6)  LIF_STEP(7)
  LIF_STEP(8)  LIF_STEP(9)  LIF_STEP(10) LIF_STEP(11)
  LIF_STEP(12) LIF_STEP(13) LIF_STEP(14) LIF_STEP(15)
  #undef LIF_STEP
}

extern "C" void kernel_launch(void* const* d_in, const int* in_sizes, int n_in,
                              void* d_out, int out_size, void* d_ws, size_t ws_size,
                              hipStream_t stream) {
  (void)in_sizes; (void)n_in; (void)out_size; (void)d_ws; (void)ws_size;
  const v4f* x = (const v4f*)d_in[0];
  v4f*       o = (v4f*)d_out;
  dim3 grid(LIF_B * (LIF_CHW4 / LIF_TPB));   // 32 * 128 = 4096 blocks
  dim3 block(LIF_TPB);                        // 256 threads = 8 wave32
  lif_fwd_kernel<<<grid, block, 0, stream>>>(x, o);
}